// ImgtoClass_Metric_81595788689922
// MI455X (gfx1250) — compile-verified
//
#include <hip/hip_runtime.h>
#include <math.h>

#define WAYN 5
#define HW   441
#define HWP  448          // padded row count (28 tiles of 16)
#define NT   28           // HWP / 16
#define NK4  160          // CCH / 4 k-steps
#define CCH  640
#define NQ   75
#define EPSV 1e-8f

typedef __attribute__((ext_vector_type(2))) float v2f;
typedef __attribute__((ext_vector_type(8))) float v8f;

// ---------------------------------------------------------------------------
// Normalize each local descriptor (L2 over C) and transpose [N,C,HW]->[N,HWP,C]
// with zero padding for rows HW..HWP-1. One block (256 threads) per padded row.
// ---------------------------------------------------------------------------
__global__ __launch_bounds__(256)
void norm_rows_kernel(const float* __restrict__ in, float* __restrict__ out) {
    int d   = blockIdx.x;            // padded descriptor id = img*HWP + s
    int img = d / HWP, s = d % HWP;
    float* dst = out + (size_t)d * CCH;
    if (s >= HW) {                   // zero pad row
        for (int c = threadIdx.x; c < CCH; c += 256) dst[c] = 0.f;
        return;
    }
    const float* base = in + (size_t)img * CCH * HW + s;
    __shared__ float red[256];
    float ss = 0.f;
    for (int c = threadIdx.x; c < CCH; c += 256) {
        float v = base[(size_t)c * HW];
        ss += v * v;
    }
    red[threadIdx.x] = ss;
    __syncthreads();
    for (int o = 128; o > 0; o >>= 1) {
        if (threadIdx.x < o) red[threadIdx.x] += red[threadIdx.x + o];
        __syncthreads();
    }
    float inv = rsqrtf(red[0]);
    for (int c = threadIdx.x; c < CCH; c += 256)
        dst[c] = base[(size_t)c * HW] * inv;
}

// ---------------------------------------------------------------------------
// sim1[i,j,s] = <supN[i,s,:], supN[j,s,:]>  (rows unit-norm -> denom = 1)
// then softmax over s. One block (256 threads) per (i,j).
// ---------------------------------------------------------------------------
__global__ __launch_bounds__(256)
void sim1_softmax_kernel(const float* __restrict__ supN, float* __restrict__ w1) {
    int i = blockIdx.x / WAYN, j = blockIdx.x % WAYN;
    __shared__ float sim[HW];
    __shared__ float red[256];
    const float* A = supN + (size_t)i * HWP * CCH;
    const float* B = supN + (size_t)j * HWP * CCH;
    for (int s = threadIdx.x; s < HW; s += 256) {
        const float* a = A + (size_t)s * CCH;
        const float* b = B + (size_t)s * CCH;
        float d = 0.f;
        for (int c = 0; c < CCH; ++c) d = fmaf(a[c], b[c], d);
        sim[s] = d;                       // denom = max(1*1, eps) = 1
    }
    __syncthreads();
    float m = -3.0e38f;
    for (int s = threadIdx.x; s < HW; s += 256) m = fmaxf(m, sim[s]);
    red[threadIdx.x] = m; __syncthreads();
    for (int o = 128; o > 0; o >>= 1) {
        if (threadIdx.x < o) red[threadIdx.x] = fmaxf(red[threadIdx.x], red[threadIdx.x + o]);
        __syncthreads();
    }
    m = red[0]; __syncthreads();
    float sum = 0.f;
    for (int s = threadIdx.x; s < HW; s += 256) {
        float e = __expf(sim[s] - m); sim[s] = e; sum += e;
    }
    red[threadIdx.x] = sum; __syncthreads();
    for (int o = 128; o > 0; o >>= 1) {
        if (threadIdx.x < o) red[threadIdx.x] += red[threadIdx.x + o];
        __syncthreads();
    }
    float inv = 1.f / red[0];
    float* dst = w1 + (size_t)blockIdx.x * HW;
    for (int s = threadIdx.x; s < HW; s += 256) dst[s] = sim[s] * inv;
}

// ---------------------------------------------------------------------------
// sim2[b,w,s] = <qN[b,s,:], supM[w,s,:]> / max(||supM[w,s]||, eps)
// (||q||=1, ||supM row|| == maskS in {0,1}); softmax over s.
// ---------------------------------------------------------------------------
__global__ __launch_bounds__(256)
void sim2_softmax_kernel(const float* __restrict__ qN, const float* __restrict__ supM,
                         const float* __restrict__ maskS, float* __restrict__ w2) {
    int b = blockIdx.x / WAYN, w = blockIdx.x % WAYN;
    __shared__ float sim[HW];
    __shared__ float red[256];
    const float* Q = qN   + (size_t)b * HWP * CCH;
    const float* S = supM + (size_t)w * HWP * CCH;
    for (int s = threadIdx.x; s < HW; s += 256) {
        const float* a  = Q + (size_t)s * CCH;
        const float* c2 = S + (size_t)s * CCH;
        float d = 0.f;
        for (int c = 0; c < CCH; ++c) d = fmaf(a[c], c2[c], d);
        sim[s] = d / fmaxf(maskS[w * HW + s], EPSV);
    }
    __syncthreads();
    float m = -3.0e38f;
    for (int s = threadIdx.x; s < HW; s += 256) m = fmaxf(m, sim[s]);
    red[threadIdx.x] = m; __syncthreads();
    for (int o = 128; o > 0; o >>= 1) {
        if (threadIdx.x < o) red[threadIdx.x] = fmaxf(red[threadIdx.x], red[threadIdx.x + o]);
        __syncthreads();
    }
    m = red[0]; __syncthreads();
    float sum = 0.f;
    for (int s = threadIdx.x; s < HW; s += 256) {
        float e = __expf(sim[s] - m); sim[s] = e; sum += e;
    }
    red[threadIdx.x] = sum; __syncthreads();
    for (int o = 128; o > 0; o >>= 1) {
        if (threadIdx.x < o) red[threadIdx.x] += red[threadIdx.x + o];
        __syncthreads();
    }
    float inv = 1.f / red[0];
    float* dst = w2 + (size_t)blockIdx.x * HW;
    for (int s = threadIdx.x; s < HW; s += 256) dst[s] = sim[s] * inv;
}

// ---------------------------------------------------------------------------
// agg[o,s] = mean_j wsoft[(o*WAYN+j), s]; thr = mean(agg) - std(agg, ddof=1);
// mask = agg > thr. Single block, 1024 threads.
// ---------------------------------------------------------------------------
__global__ __launch_bounds__(1024)
void stat_mask_kernel(const float* __restrict__ wsoft, float* __restrict__ agg,
                      float* __restrict__ mask, int nOuter) {
    const int n = nOuter * HW;
    __shared__ float redS[1024];
    __shared__ float redQ[1024];
    __shared__ float thr;
    float ls = 0.f, lq = 0.f;
    for (int idx = threadIdx.x; idx < n; idx += 1024) {
        int o = idx / HW, s = idx % HW;
        float a = 0.f;
        for (int j = 0; j < WAYN; ++j)
            a += wsoft[((size_t)(o * WAYN + j)) * HW + s];
        a *= (1.f / WAYN);
        agg[idx] = a;
        ls += a; lq += a * a;
    }
    redS[threadIdx.x] = ls; redQ[threadIdx.x] = lq;
    __syncthreads();
    for (int o = 512; o > 0; o >>= 1) {
        if (threadIdx.x < o) {
            redS[threadIdx.x] += redS[threadIdx.x + o];
            redQ[threadIdx.x] += redQ[threadIdx.x + o];
        }
        __syncthreads();
    }
    if (threadIdx.x == 0) {
        float mean = redS[0] / (float)n;
        float var  = (redQ[0] - (float)n * mean * mean) / (float)(n - 1);
        thr = mean - sqrtf(fmaxf(var, 0.f));
    }
    __syncthreads();
    for (int idx = threadIdx.x; idx < n; idx += 1024)
        mask[idx] = agg[idx] > thr ? 1.f : 0.f;
}

// ---------------------------------------------------------------------------
// data[img][s][c] *= mask[img*HW+s] for s < HW (pad rows already zero).
// Layout stride HWP.
// ---------------------------------------------------------------------------
__global__ void apply_mask_kernel(float* __restrict__ data, const float* __restrict__ mask,
                                  size_t n) {
    size_t i = (size_t)blockIdx.x * blockDim.x + threadIdx.x;
    size_t stride = (size_t)gridDim.x * blockDim.x;
    for (; i < n; i += stride) {
        size_t row = i / CCH;
        int s = (int)(row % HWP);
        size_t img = row / HWP;
        if (s < HW) data[i] *= mask[img * HW + s];
    }
}

// ---------------------------------------------------------------------------
// Pack masked support into WMMA B-fragment order:
// Bpack[(((w*NT+st)*NK4 + k4)*32 + lane)*2 + e] =
//     supM[w][st*16 + (lane&15)][4*k4 + (lane>>4) + 2*e]
// Pad rows of supM are zero, so no bounds guard needed.
// One thread per (w,st,k4,lane) -> writes a contiguous float2.
// ---------------------------------------------------------------------------
__global__ void pack_b_kernel(const float* __restrict__ supM, float* __restrict__ Bpack) {
    int idx = blockIdx.x * blockDim.x + threadIdx.x;   // 5*28*160*32 = 716800
    if (idx >= WAYN * NT * NK4 * 32) return;
    int lane = idx & 31;
    int k4   = (idx >> 5) % NK4;
    int st   = (idx / (32 * NK4)) % NT;
    int w    = idx / (32 * NK4 * NT);
    int half = lane >> 4, l15 = lane & 15;
    int row  = st * 16 + l15;
    int c    = 4 * k4 + half;
    const float* src = supM + ((size_t)w * HWP + row) * CCH;
    Bpack[(size_t)idx * 2 + 0] = src[c];
    Bpack[(size_t)idx * 2 + 1] = src[c + 2];
}

// ---------------------------------------------------------------------------
// Main kernel: per (b,w), G = Q[b] (441x640) * S[w]^T via v_wmma_f32_16x16x4_f32,
// row-wise top-3 over s, sum over rows -> out[b*5+w].
// 4 waves / block; each wave owns s-tiles wid, wid+4, ... No divergence around
// WMMA: padded rows are zero, B comes pre-packed in fragment order.
// A fragment: lane(l) = Q[qt*16 + (l&15)][k + 2*(l>>4) .. +1]   (one b64)
// B fragment: one b64 from Bpack                                 (coalesced)
// C/D: 8 VGPRs, row = v + 8*(l>>4), col = l&15
// ---------------------------------------------------------------------------
__global__ __launch_bounds__(128)
void dn4_wmma_topk_kernel(const float* __restrict__ qM, const float* __restrict__ Bpack,
                          float* __restrict__ out) {
    int b = blockIdx.x / WAYN, w = blockIdx.x % WAYN;
    const float* Qb = qM + (size_t)b * HWP * CCH;

    __shared__ float sc[16][HWP + 1];   // one q-tile's scores, padded stride
    __shared__ float t3[16][8][3];      // per-row, per-chunk top-3
    __shared__ float rowsum[16];

    const int tid  = threadIdx.x;
    const int wid  = tid >> 5;
    const int lane = tid & 31;
    const int half = lane >> 4;
    const int l15  = lane & 15;

    float accRow = 0.f;                 // per-row partial (tid < 16)

    for (int qt = 0; qt < NT; ++qt) {
        const int qrow = qt * 16 + l15;                       // < HWP, pad rows are 0
        const float* aptr = Qb + (size_t)qrow * CCH + 2 * half;

        for (int st = wid; st < NT; st += 4) {
            const float* bptr = Bpack + ((((size_t)w * NT + st) * NK4) * 32 + lane) * 2;

            v8f acc = {0.f, 0.f, 0.f, 0.f, 0.f, 0.f, 0.f, 0.f};
            #pragma unroll 8
            for (int k4 = 0; k4 < NK4; ++k4) {
                v2f a  = *(const v2f*)(aptr + 4 * k4);
                v2f bb = *(const v2f*)(bptr + 64 * k4);
                acc = __builtin_amdgcn_wmma_f32_16x16x4_f32(
                    false, a, false, bb, (short)0, acc, false, false);
            }
            const int colb = st * 16 + l15;
            #pragma unroll
            for (int v = 0; v < 8; ++v)
                sc[v + 8 * half][colb] = acc[v];
        }
        __syncthreads();

        // chunk-parallel top-3: 16 rows x 8 chunks of 56 columns
        {
            int r = tid & 15;
            int chunk = tid >> 4;
            int s0 = chunk * 56;
            int s1 = s0 + 56; if (s1 > HW) s1 = HW;   // exclude pad columns
            float t0 = -3.0e38f, t1 = -3.0e38f, t2 = -3.0e38f;
            for (int s = s0; s < s1; ++s) {
                float v = sc[r][s];
                if (v > t0)      { t2 = t1; t1 = t0; t0 = v; }
                else if (v > t1) { t2 = t1; t1 = v; }
                else if (v > t2) { t2 = v; }
            }
            t3[r][chunk][0] = t0; t3[r][chunk][1] = t1; t3[r][chunk][2] = t2;
        }
        __syncthreads();

        if (tid < 16) {
            int qg = qt * 16 + tid;
            if (qg < HW) {                               // exclude pad rows
                float t0 = -3.0e38f, t1 = -3.0e38f, t2 = -3.0e38f;
                #pragma unroll
                for (int c = 0; c < 8; ++c)
                    #pragma unroll
                    for (int e = 0; e < 3; ++e) {
                        float v = t3[tid][c][e];
                        if (v > t0)      { t2 = t1; t1 = t0; t0 = v; }
                        else if (v > t1) { t2 = t1; t1 = v; }
                        else if (v > t2) { t2 = v; }
                    }
                accRow += t0 + t1 + t2;
            }
        }
        __syncthreads();
    }

    if (tid < 16) rowsum[tid] = accRow;
    __syncthreads();
    if (tid == 0) {
        float s = 0.f;
        #pragma unroll
        for (int i = 0; i < 16; ++i) s += rowsum[i];
        out[blockIdx.x] = s;
    }
}

// ---------------------------------------------------------------------------
extern "C" void kernel_launch(void* const* d_in, const int* in_sizes, int n_in,
                              void* d_out, int out_size, void* d_ws, size_t ws_size,
                              hipStream_t stream) {
    (void)in_sizes; (void)n_in; (void)out_size; (void)ws_size;
    const float* x1 = (const float*)d_in[0];   // [75, 640, 441]
    const float* x2 = (const float*)d_in[1];   // [5, 640, 441]
    float* out = (float*)d_out;                // [75, 5]

    float* f     = (float*)d_ws;
    float* supN  = f;                                   // 5*448*640
    float* qN    = supN + (size_t)WAYN * HWP * CCH;     // 75*448*640
    float* Bpack = qN   + (size_t)NQ * HWP * CCH;       // 5*28*160*64
    float* w1    = Bpack + (size_t)WAYN * NT * NK4 * 64;// 25*441
    float* w2v   = w1   + (size_t)WAYN * WAYN * HW;     // 375*441
    float* maskS = w2v  + (size_t)NQ * WAYN * HW;       // 5*441
    float* aggS  = maskS + (size_t)WAYN * HW;           // 5*441
    float* maskQ = aggS + (size_t)WAYN * HW;            // 75*441
    float* aggQ  = maskQ + (size_t)NQ * HW;             // 75*441

    // 1) normalize + transpose into padded layout
    norm_rows_kernel<<<WAYN * HWP, 256, 0, stream>>>(x2, supN);
    norm_rows_kernel<<<NQ * HWP, 256, 0, stream>>>(x1, qN);
    // 2) support self-similarity mask
    sim1_softmax_kernel<<<WAYN * WAYN, 256, 0, stream>>>(supN, w1);
    stat_mask_kernel<<<1, 1024, 0, stream>>>(w1, aggS, maskS, WAYN);
    apply_mask_kernel<<<1024, 256, 0, stream>>>(supN, maskS, (size_t)WAYN * HWP * CCH);
    // 3) query-vs-support mask
    sim2_softmax_kernel<<<NQ * WAYN, 256, 0, stream>>>(qN, supN, maskS, w2v);
    stat_mask_kernel<<<1, 1024, 0, stream>>>(w2v, aggQ, maskQ, NQ);
    apply_mask_kernel<<<4096, 256, 0, stream>>>(qN, maskQ, (size_t)NQ * HWP * CCH);
    // 4) pack masked support into WMMA fragment order
    pack_b_kernel<<<(WAYN * NT * NK4 * 32 + 255) / 256, 256, 0, stream>>>(supN, Bpack);
    // 5) batched GEMM (fp32 WMMA) + top-3 aggregation
    dn4_wmma_topk_kernel<<<NQ * WAYN, 128, 0, stream>>>(qN, Bpack, out);
}